// VoxelPointsSampler_29334626632372
// MI455X (gfx1250) — compile-verified
//
#include <hip/hip_runtime.h>

// VoxelPointsSampler for MI455X (gfx1250, wave32).
//
// Phase A: pairwise in-radius mask via V_WMMA_F32_16X16X4_F32:
//          dis^2 = |b|^2 + |p|^2 - 2*(b.p); the cross term is a [16,2]x[2,16]
//          rank-2 GEMM per tile -> one wmma per 16x16 (box,point) tile.
//          Results packed into a [M][N/32] bitmap with wave32 ballots.
// Phase B: wave-per-box ordered first-32 selection (== top_k of binary mask
//          with lowest-index tie-break), incl. zero-mask padding, and a
//          selected-index bitmap (atomicOr) for the unique() computation.
// Phase C: single-block exclusive scan of popcounts -> rank(n) == inv value.
// Phase D: materialize uniq, then gather sampled_points / idx / query_points.

typedef __attribute__((ext_vector_type(2))) float v2f;
typedef __attribute__((ext_vector_type(8))) float v8f;

#define SAMP 32
#define GAMMA2Q 0.3025f   // (1.1 * 0.5)^2

// ---------------------------------------------------------------- init
__global__ __launch_bounds__(256) void vps_init_kernel(
    unsigned* __restrict__ sel, int* __restrict__ uniq,
    unsigned* __restrict__ ucount, int NW, int U)
{
  int t = blockIdx.x * blockDim.x + threadIdx.x;
  if (t < NW) sel[t] = 0u;
  if (t < U)  uniq[t] = 0;
  if (t == 0) *ucount = 0u;
}

// ---------------------------------------------------------------- phase A
__global__ __launch_bounds__(256) void vps_mask_kernel(
    const float* __restrict__ points, const float* __restrict__ boxes,
    unsigned* __restrict__ maskWords, int N, int M, int NW)
{
  const int lane = threadIdx.x & 31;
  const int wid  = threadIdx.x >> 5;
  const int half = lane >> 4;       // 0: lanes 0-15 (K=0,1), 1: lanes 16-31 (K=2,3 -> zeroed)
  const int l15  = lane & 15;
  const int m0   = blockIdx.x * 16; // 16-box tile

  // A matrix (boxes xy): rows m0..m0+15 in lanes 0-15; K=2,3 lanes carry zeros.
  int rowA = m0 + l15; if (rowA >= M) rowA = M - 1;
  v2f a;
  a.x = (half == 0) ? boxes[rowA * 7 + 0] : 0.0f;
  a.y = (half == 0) ? boxes[rowA * 7 + 1] : 0.0f;

  // Per-row |b|^2 and r^2, replicated to match the C/D layout
  // (VGPR v holds row v for lanes 0-15 and row v+8 for lanes 16-31).
  float bn[8], r2[8];
#pragma unroll
  for (int v = 0; v < 8; ++v) {
    int rowv = m0 + v + 8 * half;
    bool ok  = rowv < M;
    int  rc  = ok ? rowv : (M - 1);
    float bx = boxes[rc * 7 + 0], by = boxes[rc * 7 + 1];
    float dx = boxes[rc * 7 + 3], dy = boxes[rc * 7 + 4];
    bn[v] = bx * bx + by * by;
    float rr = GAMMA2Q * (dx * dx + dy * dy);
    r2[v] = ok ? rr : -1.0f;       // rows >= M can never match
  }

  const int wstart  = blockIdx.y * (blockDim.x >> 5) + wid;
  const int wstride = gridDim.y * (blockDim.x >> 5);

  for (int w = wstart; w < NW; w += wstride) {
    const int n0 = w * 32;
    int c0 = n0 + l15;       int c0c = (c0 < N) ? c0 : N - 1;
    int c1 = n0 + 16 + l15;  int c1c = (c1 < N) ? c1 : N - 1;
    float px0 = points[c0c * 5 + 0], py0 = points[c0c * 5 + 1];
    float px1 = points[c1c * 5 + 0], py1 = points[c1c * 5 + 1];
    float pn0 = px0 * px0 + py0 * py0;
    float pn1 = px1 * px1 + py1 * py1;

    // Prefetch next chunk of points (points stay L2-resident; this feeds WGP$).
    int pf = (n0 + 32 * wstride) * 5;
    if (pf < N * 5) __builtin_prefetch(&points[pf], 0, 1);

    v2f b0, b1;
    b0.x = (half == 0) ? px0 : 0.0f;  b0.y = (half == 0) ? py0 : 0.0f;
    b1.x = (half == 0) ? px1 : 0.0f;  b1.y = (half == 0) ? py1 : 0.0f;

    v8f c = {};
    v8f d0 = __builtin_amdgcn_wmma_f32_16x16x4_f32(
        false, a, false, b0, (short)0, c, false, false);
    v8f d1 = __builtin_amdgcn_wmma_f32_16x16x4_f32(
        false, a, false, b1, (short)0, c, false, false);

    // Pack compares into per-row 32-bit words via wave32 ballots.
    unsigned myword = 0u;
#define VPS_ROW(v) {                                                        \
      bool q0 = (bn[v] + pn0 - 2.0f * d0[v]) <= r2[v];                      \
      bool q1 = (bn[v] + pn1 - 2.0f * d1[v]) <= r2[v];                      \
      unsigned bal0 = __builtin_amdgcn_ballot_w32(q0);                      \
      unsigned bal1 = __builtin_amdgcn_ballot_w32(q1);                      \
      unsigned wlo = (bal0 & 0xffffu) | (bal1 << 16);                       \
      unsigned whi = (bal0 >> 16)     | (bal1 & 0xffff0000u);               \
      myword = (lane == (v))     ? wlo : myword;                            \
      myword = (lane == (v) + 8) ? whi : myword;                            \
    }
    VPS_ROW(0) VPS_ROW(1) VPS_ROW(2) VPS_ROW(3)
    VPS_ROW(4) VPS_ROW(5) VPS_ROW(6) VPS_ROW(7)
#undef VPS_ROW

    // Tail guard on point columns (no-op when N is a multiple of 32).
    if (n0 + 32 > N) {
      unsigned vm = (N > n0) ? ((1u << (N - n0)) - 1u) : 0u;
      myword &= vm;
    }
    if (lane < 16 && (m0 + lane) < M)
      maskWords[(size_t)(m0 + lane) * NW + w] = myword;
  }
}

// ---------------------------------------------------------------- phase B
__global__ __launch_bounds__(256) void vps_select_kernel(
    const unsigned* __restrict__ maskWords, unsigned* __restrict__ sel,
    int* __restrict__ sampled, int* __restrict__ hits, int N, int M, int NW)
{
  const int gw   = (blockIdx.x * blockDim.x + threadIdx.x) >> 5; // wave-per-box
  const int lane = threadIdx.x & 31;
  if (gw >= M) return;
  const unsigned* row = maskWords + (size_t)gw * NW;

  // Pass 1: first up-to-32 set bits in ascending index order.
  int found = 0;
  for (int wb = 0; wb < NW && found < SAMP; wb += 32) {
    int w = wb + lane;
    unsigned word = (w < NW) ? row[w] : 0u;
    int pc = __popc(word);
    int inc = pc;
#pragma unroll
    for (int off = 1; off < 32; off <<= 1) {
      int t = __shfl_up(inc, off, 32);
      if (lane >= off) inc += t;
    }
    int start = found + (inc - pc);
    if (word && start < SAMP) {
      unsigned x = word; int r = start;
      while (x && r < SAMP) {
        int b = __builtin_ctz(x); x &= x - 1;
        int n = w * 32 + b;
        sampled[gw * SAMP + r] = n;
        atomicOr(&sel[n >> 5], 1u << (n & 31));
        ++r;
      }
    }
    found += __shfl(inc, 31, 32);   // wave-uniform running total
  }
  int h = (found < SAMP) ? found : SAMP;
  if (lane == 0) hits[gw] = h;

  // Pass 2: pad with lowest-index zero-mask entries (these DO enter unique()).
  if (h < SAMP) {
    int need = SAMP - h;
    int zfound = 0;
    for (int wb = 0; wb < NW && zfound < need; wb += 32) {
      int w = wb + lane;
      unsigned z = 0u;
      if (w < NW) {
        z = ~row[w];
        int n0 = w * 32;
        if (n0 + 32 > N) {
          unsigned vm = (N > n0) ? ((1u << (N - n0)) - 1u) : 0u;
          z &= vm;
        }
      }
      int pc = __popc(z);
      int inc = pc;
#pragma unroll
      for (int off = 1; off < 32; off <<= 1) {
        int t = __shfl_up(inc, off, 32);
        if (lane >= off) inc += t;
      }
      int start = zfound + (inc - pc);
      if (z && start < need) {
        unsigned x = z; int r = start;
        while (x && r < need) {
          int b = __builtin_ctz(x); x &= x - 1;
          int n = w * 32 + b;
          sampled[gw * SAMP + h + r] = n;
          atomicOr(&sel[n >> 5], 1u << (n & 31));
          ++r;
        }
      }
      zfound += __shfl(inc, 31, 32);
    }
  }
}

// ---------------------------------------------------------------- phase C
__global__ __launch_bounds__(1024) void vps_scan_kernel(
    const unsigned* __restrict__ sel, unsigned* __restrict__ wprefix,
    unsigned* __restrict__ ucount, int NW)
{
  __shared__ unsigned lds[1024];
  const int tid = threadIdx.x;
  unsigned carry = 0;
  for (int base = 0; base < NW; base += 1024) {
    int w = base + tid;
    unsigned pc = (w < NW) ? (unsigned)__popc(sel[w]) : 0u;
    lds[tid] = pc;
    __syncthreads();
    for (int off = 1; off < 1024; off <<= 1) {
      unsigned t = (tid >= off) ? lds[tid - off] : 0u;
      __syncthreads();
      lds[tid] += t;
      __syncthreads();
    }
    if (w < NW) wprefix[w] = carry + lds[tid] - pc;  // exclusive
    unsigned tot = lds[1023];
    __syncthreads();
    carry += tot;
  }
  if (tid == 0) *ucount = carry;
}

__global__ __launch_bounds__(256) void vps_uniq_kernel(
    const unsigned* __restrict__ sel, const unsigned* __restrict__ wprefix,
    int* __restrict__ uniq, int NW, int cap)
{
  int w = blockIdx.x * blockDim.x + threadIdx.x;
  if (w >= NW) return;
  unsigned x = sel[w];
  unsigned base = wprefix[w];
  while (x) {
    int b = __builtin_ctz(x); x &= x - 1;
    if (base < (unsigned)cap) uniq[base] = w * 32 + b;
    ++base;
  }
}

// ---------------------------------------------------------------- phase D
__global__ __launch_bounds__(256) void vps_out_kernel(
    const float* __restrict__ points, const int* __restrict__ sampled,
    const int* __restrict__ hits, const unsigned* __restrict__ sel,
    const unsigned* __restrict__ wprefix,
    float* __restrict__ out_sp, float* __restrict__ out_idx, int M)
{
  int t = blockIdx.x * blockDim.x + threadIdx.x;
  if (t >= M * SAMP) return;
  int m = t >> 5, s = t & 31;
  bool mv = s < hits[m];
  int n = sampled[t];
  unsigned wrd   = sel[n >> 5];
  unsigned below = wrd & ((1u << (n & 31)) - 1u);
  unsigned rank  = wprefix[n >> 5] + (unsigned)__popc(below);  // == inv value
#pragma unroll
  for (int c = 0; c < 5; ++c)
    out_sp[t * 5 + c] = mv ? points[n * 5 + c] : 0.0f;
  out_idx[t] = mv ? (float)rank : 0.0f;
}

__global__ __launch_bounds__(256) void vps_query_kernel(
    const float* __restrict__ points, const int* __restrict__ uniq,
    float* __restrict__ out_q, int Q)
{
  int j = blockIdx.x * blockDim.x + threadIdx.x;
  if (j >= Q) return;
  int n = uniq[j];   // zero-padded beyond U -> points[0], matching fill_value=0
#pragma unroll
  for (int c = 0; c < 5; ++c) out_q[j * 5 + c] = points[n * 5 + c];
}

// ---------------------------------------------------------------- launch
extern "C" void kernel_launch(void* const* d_in, const int* in_sizes, int n_in,
                              void* d_out, int out_size, void* d_ws, size_t ws_size,
                              hipStream_t stream) {
  const float* points = (const float*)d_in[0];
  const float* boxes  = (const float*)d_in[1];
  (void)n_in; (void)out_size; (void)ws_size;

  const int N  = in_sizes[0] / 5;
  const int M  = in_sizes[1] / 7;
  const int NW = (N + 31) / 32;

  // Workspace layout (256B aligned); dominant piece is the [M][NW] bitmap (~25.6MB).
  auto align256 = [](size_t x) { return (x + 255) & ~(size_t)255; };
  char* ws = (char*)d_ws;
  size_t off = 0;
  unsigned* sel       = (unsigned*)(ws + off); off = align256(off + (size_t)NW * 4);
  unsigned* wprefix   = (unsigned*)(ws + off); off = align256(off + (size_t)NW * 4);
  int*      hits      = (int*)(ws + off);      off = align256(off + (size_t)M * 4);
  int*      sampled   = (int*)(ws + off);      off = align256(off + (size_t)M * SAMP * 4);
  int*      uniq      = (int*)(ws + off);      off = align256(off + (size_t)M * SAMP * 4);
  unsigned* ucount    = (unsigned*)(ws + off); off = align256(off + 256);
  unsigned* maskWords = (unsigned*)(ws + off); // M * NW u32

  float* out_sp  = (float*)d_out;                    // [M,32,5]
  float* out_idx = out_sp + (size_t)M * SAMP * 5;    // [M,32]
  float* out_q   = out_idx + (size_t)M * SAMP;       // [M*32,5]

  const int MS = M * SAMP;
  int initN = (NW > MS) ? NW : MS;
  vps_init_kernel<<<(initN + 255) / 256, 256, 0, stream>>>(sel, uniq, ucount, NW, MS);

  dim3 mgrid((M + 15) / 16, 32);
  vps_mask_kernel<<<mgrid, 256, 0, stream>>>(points, boxes, maskWords, N, M, NW);

  vps_select_kernel<<<(MS + 255) / 256, 256, 0, stream>>>(
      maskWords, sel, sampled, hits, N, M, NW);

  vps_scan_kernel<<<1, 1024, 0, stream>>>(sel, wprefix, ucount, NW);

  vps_uniq_kernel<<<(NW + 255) / 256, 256, 0, stream>>>(sel, wprefix, uniq, NW, MS);

  vps_out_kernel<<<(MS + 255) / 256, 256, 0, stream>>>(
      points, sampled, hits, sel, wprefix, out_sp, out_idx, M);

  vps_query_kernel<<<(MS + 255) / 256, 256, 0, stream>>>(points, uniq, out_q, MS);
}